// CUDAMSDeformableAttention_2860448219655
// MI455X (gfx1250) — compile-verified
//
#include <hip/hip_runtime.h>
#include <hip/hip_bf16.h>
#include <math.h>

// ---------------- problem constants (fixed by the reference) ----------------
#define EMBED    256
#define HEADS    8
#define HEAD_DIM 32
#define LEVELS   4
#define POINTS   4
#define BSZ      2
#define NQ       10000
#define NVAL     21760   // 128*128 + 64*64 + 32*32 + 16*16
#define KDIM     256     // K is 256 in every GEMM here

__constant__ int LVL_H_c[4]     = {128, 64, 32, 16};
__constant__ int LVL_W_c[4]     = {128, 64, 32, 16};
__constant__ int LVL_START_c[4] = {0, 16384, 20480, 21504};

typedef float v2f __attribute__((ext_vector_type(2)));
typedef float v8f __attribute__((ext_vector_type(8)));

// ---------------------------------------------------------------------------
// FP32 WMMA GEMM:  C[M,N] = A[M,K] @ B[K,N] + bias[N] (+ residual[M,N])
//
// Block = 256 threads (8 waves). Block tile = (8 m-tiles x 16 rows) x 64 cols.
// The 64-column B strip (K x 64 = 64KB) is staged in LDS once per block via
// CDNA5 async global->LDS loads (ASYNCcnt), with a pair-interleaved,
// XOR-32-swizzled layout so each wave's B fragment is one conflict-free
// ds_load_b64. A fragments are software-pipelined global b64 loads.
// gridDim = (N/64, ceil(M/16/8)). EXEC is all-ones at every WMMA.
// ---------------------------------------------------------------------------
__global__ __launch_bounds__(256)
void wmma_gemm_f32(const float* __restrict__ A,
                   const float* __restrict__ B,
                   const float* __restrict__ bias,
                   const float* __restrict__ residual,
                   float* __restrict__ C,
                   int M, int N)
{
    // LDS B slab: 128 K-pairs * 128 floats (64 cols * 2 K-values) = 64KB
    __shared__ float Bsh[128 * 128];

    const int tid  = threadIdx.x;
    const int lane = tid & 31;
    const int wave = tid >> 5;
    const int n0   = blockIdx.x << 6;

    // ---- cooperative async fill of the B strip (K x 64) into LDS ----------
    // element B[k][c] -> Bsh[(k>>1)*128 + (((c<<1)|(k&1)) ^ (((k>>1)&1)<<5))]
    {
        const float* Bgp = B + n0;
        #pragma unroll 8
        for (int i = 0; i < 64; ++i) {
            const int e  = (i << 8) + tid;    // 0..16383
            const int k  = e >> 6;            // 0..255
            const int c  = e & 63;            // 0..63
            const int kp = k >> 1;
            const int idx = kp * 128 + (((c << 1) | (k & 1)) ^ ((kp & 1) << 5));
            const unsigned ldsaddr = (unsigned)(unsigned long long)(&Bsh[idx]);
            const float* src = Bgp + ((size_t)k * N + c);
            asm volatile("global_load_async_to_lds_b32 %0, %1, off"
                         :: "v"(ldsaddr), "v"(src) : "memory");
        }
        asm volatile("s_wait_asynccnt 0x0" ::: "memory");
    }
    __syncthreads();

    const int mt = blockIdx.y * 8 + wave;     // this wave's 16-row m-tile
    if (mt >= (M >> 4)) return;               // wave-uniform exit (after barrier)

    const int m0 = mt << 4;
    const int lr = lane & 15;                 // row / col within half-wave
    const int hi = lane >> 4;                 // K-half / M-half selector

    v8f c0 = {}, c1 = {}, c2 = {}, c3 = {};

    const float* Arow = A + (size_t)(m0 + lr) * KDIM;

    // A fragment for k=0 (lanes0-15: K=k,k+1 ; lanes16-31: K=k+2,k+3)
    v2f a_cur = *(const v2f*)&Arow[2 * hi];

    for (int k = 0; k < KDIM; k += 4) {
        // software-pipelined A prefetch for next k-step
        const int kn = (k + 4 < KDIM) ? (k + 4) : k;
        const v2f a_nxt = *(const v2f*)&Arow[kn + 2 * hi];

        // B fragments from LDS: pair row kp = k/2 + hi, col = lr + 16*t
        const int kp = (k >> 1) + hi;
        const int sw = (kp & 1) << 5;
        const float* Bl = &Bsh[kp * 128];
        const v2f b0 = *(const v2f*)&Bl[((lr      ) << 1) ^ sw];
        const v2f b1 = *(const v2f*)&Bl[((lr + 16 ) << 1) ^ sw];
        const v2f b2 = *(const v2f*)&Bl[((lr + 32 ) << 1) ^ sw];
        const v2f b3 = *(const v2f*)&Bl[((lr + 48 ) << 1) ^ sw];

        c0 = __builtin_amdgcn_wmma_f32_16x16x4_f32(false, a_cur, false, b0, (short)0, c0, false, false);
        c1 = __builtin_amdgcn_wmma_f32_16x16x4_f32(false, a_cur, false, b1, (short)0, c1, false, false);
        c2 = __builtin_amdgcn_wmma_f32_16x16x4_f32(false, a_cur, false, b2, (short)0, c2, false, false);
        c3 = __builtin_amdgcn_wmma_f32_16x16x4_f32(false, a_cur, false, b3, (short)0, c3, false, false);

        a_cur = a_nxt;
    }

    // C/D layout: VGPR i -> M = m0 + i + 8*hi, N = n0 + lr (+16t)
    #pragma unroll
    for (int i = 0; i < 8; ++i) {
        const int row   = m0 + i + 8 * hi;
        const size_t rb = (size_t)row * N;
        const int col   = n0 + lr;

        float r0 = c0[i] + bias[col];
        float r1 = c1[i] + bias[col + 16];
        float r2 = c2[i] + bias[col + 32];
        float r3 = c3[i] + bias[col + 48];
        if (residual) {
            r0 += residual[rb + col];
            r1 += residual[rb + col + 16];
            r2 += residual[rb + col + 32];
            r3 += residual[rb + col + 48];
        }
        C[rb + col]      = r0;
        C[rb + col + 16] = r1;
        C[rb + col + 32] = r2;
        C[rb + col + 48] = r3;
    }
}

// ---------------------------------------------------------------------------
// Deformable sampling core.
// Grid: BS*Q blocks, 256 threads = 8 waves; wave = head, lane = head dim.
// Per (b,q,h): softmax over 16 (level,point) logits via wave shuffles, then
// 16 bilinear samples (4 corner gathers each, 128B-coalesced across lanes).
// ---------------------------------------------------------------------------
__global__ __launch_bounds__(256)
void msda_sample(const float* __restrict__ vproj,   // (BS, NV, HEADS, HEAD_DIM)
                 const float* __restrict__ off,     // (BS*Q, 256)
                 const float* __restrict__ logits,  // (BS*Q, 128)
                 const float* __restrict__ refp,    // (BS, Q, LEVELS, 2)
                 float* __restrict__ mid)           // (BS*Q, 256)
{
    const int bq   = blockIdx.x;
    const int b    = bq / NQ;
    const int h    = threadIdx.x >> 5;
    const int lane = threadIdx.x & 31;

    // ---- softmax over 16 attention logits (lanes 0..15 hold one each) ----
    float logit = -INFINITY;
    if (lane < 16) logit = logits[(size_t)bq * (HEADS * LEVELS * POINTS) + h * 16 + lane];
    float mx = logit;
    #pragma unroll
    for (int d = 8; d >= 1; d >>= 1) mx = fmaxf(mx, __shfl_xor(mx, d, 16));
    float ex = (lane < 16) ? __expf(logit - mx) : 0.0f;
    float sm = ex;
    #pragma unroll
    for (int d = 8; d >= 1; d >>= 1) sm += __shfl_xor(sm, d, 16);
    const float wsm = ex / sm;   // valid in lanes 0..15 (sources for __shfl below)

    float acc = 0.0f;

    #pragma unroll
    for (int l = 0; l < LEVELS; ++l) {
        const int W = LVL_W_c[l];
        const int H = LVL_H_c[l];
        const float rx = refp[((size_t)bq * LEVELS + l) * 2 + 0];
        const float ry = refp[((size_t)bq * LEVELS + l) * 2 + 1];
        const float* vbase =
            vproj + (((size_t)b * NVAL + LVL_START_c[l]) * HEADS + h) * HEAD_DIM + lane;

        #pragma unroll
        for (int p = 0; p < POINTS; ++p) {
            const size_t ob = (size_t)bq * EMBED + (((h * LEVELS + l) * POINTS + p) << 1);
            const float ox = off[ob + 0];
            const float oy = off[ob + 1];

            const float x = (rx + ox / (float)W) * (float)W - 0.5f;
            const float y = (ry + oy / (float)H) * (float)H - 0.5f;

            const float x0f = floorf(x), y0f = floorf(y);
            const float lx = x - x0f, ly = y - y0f;
            const int   x0 = (int)x0f, y0 = (int)y0f;

            const float aw = __shfl(wsm, l * POINTS + p, 32);

            #pragma unroll
            for (int c = 0; c < 4; ++c) {
                const int dy = c >> 1, dx = c & 1;
                const float cw = (dy ? ly : 1.0f - ly) * (dx ? lx : 1.0f - lx);
                const int xi = x0 + dx, yi = y0 + dy;
                const bool valid = (xi >= 0) & (xi < W) & (yi >= 0) & (yi < H);
                const int xc = min(max(xi, 0), W - 1);
                const int yc = min(max(yi, 0), H - 1);
                const size_t idx = (size_t)(yc * W + xc);
                const float val = vbase[idx * (HEADS * HEAD_DIM)];
                acc += val * (aw * cw * (valid ? 1.0f : 0.0f));
            }
        }
    }

    mid[(size_t)bq * EMBED + h * HEAD_DIM + lane] = acc;
}

// ---------------------------------------------------------------------------
extern "C" void kernel_launch(void* const* d_in, const int* in_sizes, int n_in,
                              void* d_out, int out_size, void* d_ws, size_t ws_size,
                              hipStream_t stream)
{
    const float* query  = (const float*)d_in[0];   // (BS, Q, 256)
    const float* value  = (const float*)d_in[1];   // (BS, NV, 256)
    const float* refp   = (const float*)d_in[2];   // (BS, Q, LEVELS, 2)
    // d_in[3] spatial_shapes: constant, baked in
    const float* W_off  = (const float*)d_in[4];   // (256, 256)
    const float* b_off  = (const float*)d_in[5];   // (256,)
    const float* W_attn = (const float*)d_in[6];   // (256, 128)
    const float* b_attn = (const float*)d_in[7];   // (128,)
    const float* W_val  = (const float*)d_in[8];   // (256, 256)
    const float* b_val  = (const float*)d_in[9];   // (256,)
    const float* W_out  = (const float*)d_in[10];  // (256, 256)
    const float* b_out  = (const float*)d_in[11];  // (256,)
    float* out = (float*)d_out;

    // workspace layout (floats)
    float* ws     = (float*)d_ws;
    float* vproj  = ws;                                         // BS*NV*256
    float* offw   = vproj  + (size_t)BSZ * NVAL * EMBED;        // BS*Q*256
    float* logitw = offw   + (size_t)BSZ * NQ * EMBED;          // BS*Q*128
    float* midw   = logitw + (size_t)BSZ * NQ * 128;            // BS*Q*256

    const int THREADS = 256;

    // 1) v = value @ W_val + b_val            M=43520, N=256
    {
        int M = BSZ * NVAL, N = EMBED;
        dim3 grid(N / 64, ((M / 16) + 7) / 8);
        wmma_gemm_f32<<<grid, THREADS, 0, stream>>>(value, W_val, b_val, nullptr, vproj, M, N);
    }
    // 2) off = query @ W_off + b_off          M=20000, N=256
    {
        int M = BSZ * NQ, N = EMBED;
        dim3 grid(N / 64, ((M / 16) + 7) / 8);
        wmma_gemm_f32<<<grid, THREADS, 0, stream>>>(query, W_off, b_off, nullptr, offw, M, N);
    }
    // 3) logits = query @ W_attn + b_attn     M=20000, N=128
    {
        int M = BSZ * NQ, N = 128;
        dim3 grid(N / 64, ((M / 16) + 7) / 8);
        wmma_gemm_f32<<<grid, THREADS, 0, stream>>>(query, W_attn, b_attn, nullptr, logitw, M, N);
    }
    // 4) deformable bilinear sampling + attention-weighted sum
    msda_sample<<<BSZ * NQ, THREADS, 0, stream>>>(vproj, offw, logitw, refp, midw);

    // 5) out = mid @ W_out + b_out + query    M=20000, N=256
    {
        int M = BSZ * NQ, N = EMBED;
        dim3 grid(N / 64, ((M / 16) + 7) / 8);
        wmma_gemm_f32<<<grid, THREADS, 0, stream>>>(midw, W_out, b_out, query, out, M, N);
    }
}